// PyramidOccupancyNetwork_26903675142681
// MI455X (gfx1250) — compile-verified
//
#include <hip/hip_runtime.h>

// MI455X (gfx1250, wave32) implementation of aggregate_features:
//   out[i] = sum_j adj[i,j] * bilinear_warp(x[j], Minv[i,j])
//
// Phase 1: analytic per-pair inverse-affine (transforms are rigid SE(2)).
// Phase 2: V_WMMA_F32_16X16X4_F32 computes (gx,gy) for 16 pixels x 8 pairs
//          per wave in a single matrix op:  D[16x16] = coords[16x4] x MinvT[4x16].
// Phase 3: gather-bound bilinear sampling, 16 channel accumulators per lane,
//          clamped+masked taps (exactly the reference's clip*valid semantics).

typedef float v2f __attribute__((ext_vector_type(2)));
typedef float v8f __attribute__((ext_vector_type(8)));

#define N_AG   8
#define CH     32
#define IMG_H  134
#define IMG_W  103
#define HW     (IMG_H * IMG_W)        // 13802
#define PPM_F  6.7f
#define CX_F   51.5f
#define CY_F   67.0f
#define WPB    8                      // waves per block
#define TILES  ((HW + 15) / 16)       // 863 16-pixel tiles

__global__ __launch_bounds__(256)
void warp_aggregate_kernel(const float* __restrict__ x,
                           const float* __restrict__ tf,
                           const int*   __restrict__ adj,
                           float*       __restrict__ out)
{
    __shared__ float sMinv[N_AG][6];            // [j][{i00,i01,io0,i10,i11,io1}]
    __shared__ float sCoord[WPB][16][16];       // [wave][pixel][col]; col 2j=gx, 2j+1=gy

    const int i    = blockIdx.y;
    const int tid  = threadIdx.x;
    const int lane = tid & 31;
    const int wave = tid >> 5;

    // ---------- Phase 1: Minv for pairs (i, j=0..7) ----------
    if (tid < N_AG) {
        const int j = tid;
        const float* Ti = tf + i * 16;
        const float* Tj = tf + j * 16;
        float ai00 = Ti[0], ai01 = Ti[1], ti0 = Ti[3];
        float ai10 = Ti[4], ai11 = Ti[5], ti1 = Ti[7];
        float aj00 = Tj[0], aj01 = Tj[1], tj0 = Tj[3];
        float aj10 = Tj[4], aj11 = Tj[5], tj1 = Tj[7];
        // rel = inv(Ti) * Tj; Ti rigid => inv rotation = Ai^T, inv trans = -Ai^T ti
        float R00 = ai00*aj00 + ai10*aj10;
        float R01 = ai00*aj01 + ai10*aj11;
        float R10 = ai01*aj00 + ai11*aj10;
        float R11 = ai01*aj01 + ai11*aj11;
        float dx = tj0 - ti0, dy = tj1 - ti1;
        float tx = (ai00*dx + ai10*dy) * PPM_F;
        float ty = (ai01*dx + ai11*dy) * PPM_F;
        float off0 = CX_F - (R00*CX_F + R01*CY_F) + tx;
        float off1 = CY_F - (R10*CX_F + R11*CY_F) + ty;
        // invert [[R, off],[0,0,1]]
        float det  = R00*R11 - R01*R10;
        float rdet = 1.0f / det;
        float i00 =  R11*rdet, i01 = -R01*rdet;
        float i10 = -R10*rdet, i11 =  R00*rdet;
        float io0 = -(i00*off0 + i01*off1);
        float io1 = -(i10*off0 + i11*off1);
        sMinv[j][0] = i00; sMinv[j][1] = i01; sMinv[j][2] = io0;
        sMinv[j][3] = i10; sMinv[j][4] = i11; sMinv[j][5] = io1;
    }
    __syncthreads();

    // ---------- Phase 2: WMMA coordinate generation ----------
    const int  tile      = blockIdx.x * WPB + wave;
    const int  tile_base = tile * 16;
    const bool hi        = (lane >= 16);
    const int  m         = lane & 15;           // pixel-in-tile / column index

    // A 16x4 f32 layout: lanes 0-15 hold K=0,1 (x,y); lanes 16-31 hold K=2,3 (1,0)
    v2f a;
    {
        int p = tile_base + m;
        if (p >= HW) p = HW - 1;                // clamp: OOB pixels never stored
        float fx = (float)(p % IMG_W);
        float fy = (float)(p / IMG_W);
        a.x = hi ? 1.0f : fx;
        a.y = hi ? 0.0f : fy;
    }
    // B 4x16 f32 layout: lanes 0-15 hold K=0,1; lanes 16-31 hold K=2,3 (coeff of '1', and 0)
    v2f b;
    {
        int j = m >> 1;
        int r = m & 1;                          // 0: gx row, 1: gy row
        const float* mv = &sMinv[j][r * 3];
        b.x = hi ? mv[2] : mv[0];
        b.y = hi ? 0.0f  : mv[1];
    }
    v8f cacc = {};
    v8f dmat = __builtin_amdgcn_wmma_f32_16x16x4_f32(
        /*neg_a=*/false, a, /*neg_b=*/false, b,
        /*c_mod=*/(short)0, cacc, /*reuse_a=*/false, /*reuse_b=*/false);

    // D layout: vgpr r, lanes 0-15 -> (M=r, N=lane); lanes 16-31 -> (M=r+8, N=lane-16)
    #pragma unroll
    for (int r = 0; r < 8; ++r) {
        int pix = r + (hi ? 8 : 0);
        sCoord[wave][pix][m] = dmat[r];
    }
    __syncthreads();

    // ---------- Phase 3: bilinear gather + masked sum ----------
    const int p   = tile_base + m;              // this lane's pixel
    const int chi = hi ? 1 : 0;                 // channel parity for this lane
    float acc[16];
    #pragma unroll
    for (int k = 0; k < 16; ++k) acc[k] = 0.0f;

    for (int j = 0; j < N_AG; ++j) {
        if (adj[i * N_AG + j] == 0) continue;   // uniform branch (scalar load)
        float gx = sCoord[wave][m][2*j];
        float gy = sCoord[wave][m][2*j + 1];
        float x0f = floorf(gx), y0f = floorf(gy);
        float wx = gx - x0f, wy = gy - y0f;
        int x0 = (int)x0f, y0 = (int)y0f;
        int x1 = x0 + 1,   y1 = y0 + 1;
        float v00 = (x0 >= 0 && x0 < IMG_W && y0 >= 0 && y0 < IMG_H) ? 1.0f : 0.0f;
        float v01 = (x1 >= 0 && x1 < IMG_W && y0 >= 0 && y0 < IMG_H) ? 1.0f : 0.0f;
        float v10 = (x0 >= 0 && x0 < IMG_W && y1 >= 0 && y1 < IMG_H) ? 1.0f : 0.0f;
        float v11 = (x1 >= 0 && x1 < IMG_W && y1 >= 0 && y1 < IMG_H) ? 1.0f : 0.0f;
        float pw00 = (1.f - wx) * (1.f - wy) * v00;
        float pw01 = wx * (1.f - wy) * v01;
        float pw10 = (1.f - wx) * wy * v10;
        float pw11 = wx * wy * v11;
        int xc0 = min(max(x0, 0), IMG_W - 1), xc1 = min(max(x1, 0), IMG_W - 1);
        int yc0 = min(max(y0, 0), IMG_H - 1), yc1 = min(max(y1, 0), IMG_H - 1);
        int r0 = yc0 * IMG_W, r1 = yc1 * IMG_W;
        const float* img = x + ((size_t)j * CH + chi) * HW;
        #pragma unroll
        for (int k = 0; k < 16; ++k) {
            const float* ch = img + (size_t)(2 * k) * HW;
            acc[k] += pw00 * ch[r0 + xc0] + pw01 * ch[r0 + xc1]
                    + pw10 * ch[r1 + xc0] + pw11 * ch[r1 + xc1];
        }
    }

    if (p < HW) {
        float* ob = out + ((size_t)i * CH + chi) * HW + p;
        #pragma unroll
        for (int k = 0; k < 16; ++k) ob[(size_t)(2 * k) * HW] = acc[k];
    }
}

extern "C" void kernel_launch(void* const* d_in, const int* in_sizes, int n_in,
                              void* d_out, int out_size, void* d_ws, size_t ws_size,
                              hipStream_t stream) {
    (void)in_sizes; (void)n_in; (void)d_ws; (void)ws_size; (void)out_size;
    const float* x   = (const float*)d_in[0];   // [8,32,134,103] f32
    const float* tf  = (const float*)d_in[1];   // [8,4,4] f32
    const int*   adj = (const int*)d_in[2];     // [8,8] i32
    float*       out = (float*)d_out;           // [8,32,134,103] f32

    dim3 grid((TILES + WPB - 1) / WPB, N_AG);   // (108, 8)
    warp_aggregate_kernel<<<grid, 256, 0, stream>>>(x, tf, adj, out);
}